// Mymodel1_42073499632165
// MI455X (gfx1250) — compile-verified
//
#include <hip/hip_runtime.h>
#include <hip/hip_bf16.h>

typedef float v2f __attribute__((ext_vector_type(2)));
typedef float v8f __attribute__((ext_vector_type(8)));

namespace {
constexpr int kB = 64;
constexpr int kT = 1024;
constexpr int kE = 2048;
constexpr int kSOut = kE - 1;                                   // 2047 diagonals out
constexpr int kDiagsPerWave  = 16;
constexpr int kWavesPerBlock = 8;
constexpr int kDiagsPerBlock = kDiagsPerWave * kWavesPerBlock;  // 128
constexpr int kStep = kE - 1;                                   // index delta per row along a diagonal
}

// One wave owns 16 anti-diagonals (s = sbase..sbase+15); lane n owns diagonal
// sbase+n. Element (r, s-r) has flat index s + r*(E-1): one running int offset
// per lane, sibling rows as compile-time immediates. B-fragment (4x16 f32):
// VGPR0 = rows K{0,2} across half-waves, VGPR1 = rows K{1,3}. D = ones(16x4) x
// B + C accumulates column sums; c[0] in lanes 0..15 = the 16 diagonal sums.
__global__ __launch_bounds__(256)
void hankel_diag_mean_wmma(const float* __restrict__ in, float* __restrict__ out) {
  const int lane = threadIdx.x & 31;
  const int wave = __builtin_amdgcn_readfirstlane(threadIdx.x >> 5);  // SGPR
  const int n    = lane & 15;  // column (diagonal) within the wave's 16-group
  const int half = lane >> 4;  // 0 -> B rows K={0,1}; 1 -> B rows K={2,3}
  const int batch = blockIdx.y;
  const int sbase = kT + blockIdx.x * kDiagsPerBlock + wave * kDiagsPerWave;  // scalar
  const int s     = sbase + n;                 // this lane's diagonal: i + j = s

  const float* __restrict__ src = in + (size_t)batch * kT * kE;

  v8f c = {};
  v2f a; a[0] = 1.0f; a[1] = 1.0f;             // A = ones(16x4), all lanes

  // First contributing row for this group (skip rows where every lane's j = s - r exceeds E-1).
  int i0 = sbase - (kE - 1);
  if (i0 < 0) i0 = 0;
  i0 &= ~15;
  // First row from which NO lane needs the j <= E-1 clamp:
  // r >= sbase + 15 - (E-1) for every lane/K. Align up to 16.
  int iClean = (sbase - (kE - 1) + 15 + 15) & ~15;
  if (iClean < i0) iClean = i0;
  if (iClean > kT) iClean = kT;

  // --- Head: ragged upper edge, clamp + zero out-of-range lanes (few iters).
  for (int i = i0; i < iClean; i += 4) {
    const int r0 = i + 2 * half;
    const int r1 = r0 + 1;
    const int j0 = s - r0;                     // j >= 1 always (s >= 1024 > r)
    const int j1 = s - r1;
    const int jc0 = j0 < (kE - 1) ? j0 : (kE - 1);
    const int jc1 = j1 < (kE - 1) ? j1 : (kE - 1);
    const float x0 = __builtin_nontemporal_load(src + r0 * kE + jc0);
    const float x1 = __builtin_nontemporal_load(src + r1 * kE + jc1);
    v2f b;
    b[0] = (j0 < kE) ? x0 : 0.0f;              // EXEC stays all-ones for WMMA
    b[1] = (j1 < kE) ? x1 : 0.0f;
    c = __builtin_amdgcn_wmma_f32_16x16x4_f32(false, a, false, b, (short)0, c,
                                              false, false);
  }

  // --- Steady state: 16 rows/iter, 8 loads in one clause, 4 WMMAs, no clamps.
  int idx = s + (iClean + 2 * half) * kStep;   // flat index of (row iClean+2h, diag s)
  for (int i = iClean; i < kT; i += 16) {
    const float x0 = __builtin_nontemporal_load(src + idx);               // row i    + 2h
    const float x1 = __builtin_nontemporal_load(src + idx +  1 * kStep);  // row i+1  + 2h
    const float x2 = __builtin_nontemporal_load(src + idx +  4 * kStep);  // row i+4  + 2h
    const float x3 = __builtin_nontemporal_load(src + idx +  5 * kStep);  // row i+5  + 2h
    const float x4 = __builtin_nontemporal_load(src + idx +  8 * kStep);  // row i+8  + 2h
    const float x5 = __builtin_nontemporal_load(src + idx +  9 * kStep);  // row i+9  + 2h
    const float x6 = __builtin_nontemporal_load(src + idx + 12 * kStep);  // row i+12 + 2h
    const float x7 = __builtin_nontemporal_load(src + idx + 13 * kStep);  // row i+13 + 2h
    v2f b0; b0[0] = x0; b0[1] = x1;
    v2f b1; b1[0] = x2; b1[1] = x3;
    v2f b2; b2[0] = x4; b2[1] = x5;
    v2f b3; b3[0] = x6; b3[1] = x7;
    c = __builtin_amdgcn_wmma_f32_16x16x4_f32(false, a, false, b0, (short)0, c,
                                              false, false);
    c = __builtin_amdgcn_wmma_f32_16x16x4_f32(false, a, false, b1, (short)0, c,
                                              false, false);
    c = __builtin_amdgcn_wmma_f32_16x16x4_f32(false, a, false, b2, (short)0, c,
                                              false, false);
    c = __builtin_amdgcn_wmma_f32_16x16x4_f32(false, a, false, b3, (short)0, c,
                                              false, false);
    idx += 16 * kStep;
  }

  // C/D layout: VGPR0, lanes 0..15 = D[M=0, N=lane] = sum of diagonal sbase+lane.
  const int d = s - kT;
  if (half == 0 && d < kSOut) {
    const float cnt = (float)((s <= kE - 1) ? kT : (kT + kE - 1 - s));
    out[(size_t)batch * kSOut + d] = c[0] / cnt;
  }
}

extern "C" void kernel_launch(void* const* d_in, const int* in_sizes, int n_in,
                              void* d_out, int out_size, void* d_ws, size_t ws_size,
                              hipStream_t stream) {
  (void)in_sizes; (void)n_in; (void)out_size; (void)d_ws; (void)ws_size;
  const float* x = (const float*)d_in[0];
  float* y = (float*)d_out;
  dim3 grid((kSOut + kDiagsPerBlock - 1) / kDiagsPerBlock, kB);  // (16, 64)
  dim3 block(kWavesPerBlock * 32);                               // 8 waves of 32
  hipLaunchKernelGGL(hankel_diag_mean_wmma, grid, block, 0, stream, x, y);
}